// ChebEncoder2_82781199663548
// MI455X (gfx1250) — compile-verified
//
#include <hip/hip_runtime.h>

typedef float v2f __attribute__((ext_vector_type(2)));
typedef float v8f __attribute__((ext_vector_type(8)));

#define FDIM 64

// ---------------- utility ----------------
__global__ void k_zero(float* __restrict__ p, long n) {
  long i = (long)blockIdx.x * blockDim.x + threadIdx.x;
  long stride = (long)gridDim.x * blockDim.x;
  for (; i < n; i += stride) p[i] = 0.0f;
}

// deg[row[e]] += 1
__global__ void k_deg(const int* __restrict__ row, float* __restrict__ deg, int E) {
  int i = blockIdx.x * blockDim.x + threadIdx.x;
  int stride = gridDim.x * blockDim.x;
  for (; i < E; i += stride) atomicAdd(&deg[row[i]], 1.0f);
}

// in-place: deg -> deg^-1/2 (0 if deg==0)
__global__ void k_dis(float* __restrict__ deg, int N) {
  int i = blockIdx.x * blockDim.x + threadIdx.x;
  int stride = gridDim.x * blockDim.x;
  for (; i < N; i += stride) {
    float d = deg[i];
    deg[i] = (d > 0.0f) ? rsqrtf(d) : 0.0f;
  }
}

// w[e] = -dis[row]*dis[col]
__global__ void k_edgew(const int* __restrict__ row, const int* __restrict__ col,
                        const float* __restrict__ dis, float* __restrict__ w, int E) {
  int i = blockIdx.x * blockDim.x + threadIdx.x;
  int stride = gridDim.x * blockDim.x;
  for (; i < E; i += stride) w[i] = -dis[row[i]] * dis[col[i]];
}

// T[col[e], f] += w[e] * X[row[e], f]
// one thread per (edge, 4-feature chunk): indices loaded once per 16 threads,
// gather is a single global_load_b128.
__global__ void k_scatter(const int* __restrict__ row, const int* __restrict__ col,
                          const float* __restrict__ w, const float* __restrict__ X,
                          float* __restrict__ T, long E16) {
  long i = (long)blockIdx.x * blockDim.x + threadIdx.x;
  long stride = (long)gridDim.x * blockDim.x;
  for (; i < E16; i += stride) {
    long e = i >> 4;
    int f4 = (int)(i & 15) << 2;
    int r = row[e];
    int c = col[e];
    float we = w[e];
    const float4 xv = *(const float4*)(X + (long)r * FDIM + f4);
    float* tp = T + (long)c * FDIM + f4;
    atomicAdd(tp + 0, we * xv.x);
    atomicAdd(tp + 1, we * xv.y);
    atomicAdd(tp + 2, we * xv.z);
    atomicAdd(tp + 3, we * xv.w);
  }
}

// OUT = X @ W[0] + T @ W[1] + bias   using full-fp32 WMMA (16x16x4).
// W (2,64,64) is staged in LDS once per block; one wave owns a 16x64 tile.
__global__ __launch_bounds__(256)
void k_cheb_gemm(const float* __restrict__ X, const float* __restrict__ T,
                 const float* __restrict__ W, const float* __restrict__ bias,
                 float* __restrict__ OUT, int N) {
  __shared__ float Wsh[2 * FDIM * FDIM];  // 32 KB

  // cooperative coalesced staging (b128)
  {
    const float4* __restrict__ src = (const float4*)W;
    float4* dst = (float4*)Wsh;
#pragma unroll
    for (int i = 0; i < (2 * FDIM * FDIM / 4) / 256; ++i)
      dst[threadIdx.x + i * 256] = src[threadIdx.x + i * 256];
  }
  __syncthreads();  // before any wave exits

  const int lane = threadIdx.x & 31;
  const int wave = threadIdx.x >> 5;
  const int rowbase = blockIdx.x * 128 + wave * 16;
  if (rowbase >= N) return;            // uniform per-wave branch, EXEC stays all-1s

  const int m    = lane & 15;          // A-fragment row  (ISA 7.12.2: 32-bit A 16x4)
  const int kb   = (lane >> 4) << 1;   // K base: 0 for lanes 0-15, 2 for lanes 16-31
  const int ncol = lane & 15;          // B/C-fragment column

  int arow = rowbase + m;
  if (arow > N - 1) arow = N - 1;      // clamp loads; stores are predicated below
  const float* __restrict__ xrow = X + (long)arow * FDIM;
  const float* __restrict__ trow = T + (long)arow * FDIM;

  v8f acc[4];
#pragma unroll
  for (int nt = 0; nt < 4; ++nt) acc[nt] = (v8f){0, 0, 0, 0, 0, 0, 0, 0};

  for (int k0 = 0; k0 < FDIM; k0 += 4) {
    v2f ax, at;
    ax[0] = xrow[k0 + kb];
    ax[1] = xrow[k0 + kb + 1];
    at[0] = trow[k0 + kb];
    at[1] = trow[k0 + kb + 1];
    const float* w0k = Wsh + (k0 + kb) * FDIM;
    const float* w1k = Wsh + FDIM * FDIM + (k0 + kb) * FDIM;
#pragma unroll
    for (int nt = 0; nt < 4; ++nt) {
      v2f b0, b1v;
      b0[0]  = w0k[nt * 16 + ncol];
      b0[1]  = w0k[FDIM + nt * 16 + ncol];
      acc[nt] = __builtin_amdgcn_wmma_f32_16x16x4_f32(
          false, ax, false, b0, (short)0, acc[nt], false, false);
      b1v[0] = w1k[nt * 16 + ncol];
      b1v[1] = w1k[FDIM + nt * 16 + ncol];
      acc[nt] = __builtin_amdgcn_wmma_f32_16x16x4_f32(
          false, at, false, b1v, (short)0, acc[nt], false, false);
    }
  }

  // C/D layout: VGPR r holds M = r + 8*(lane>=16), N = lane&15
  const int mhalf = (lane >> 4) * 8;
#pragma unroll
  for (int r = 0; r < 8; ++r) {
    int orow = rowbase + mhalf + r;
    if (orow < N) {
#pragma unroll
      for (int nt = 0; nt < 4; ++nt)
        OUT[(long)orow * FDIM + nt * 16 + ncol] = acc[nt][r] + bias[nt * 16 + ncol];
    }
  }
}

// BatchNorm partial stats: per-block local accumulation then 2 atomics per thread
__global__ void k_bnstats(const float* __restrict__ h, float* __restrict__ sum,
                          float* __restrict__ sumsq, int N) {
  const int f = threadIdx.x & 63;
  const int sub = threadIdx.x >> 6;  // 0..3
  long r = (long)blockIdx.x * 256 + sub;
  long rend = (long)(blockIdx.x + 1) * 256;
  if (rend > N) rend = N;
  float s = 0.0f, sq = 0.0f;
  for (; r < rend; r += 4) {
    float v = h[r * FDIM + f];
    s += v;
    sq += v * v;
  }
  atomicAdd(&sum[f], s);
  atomicAdd(&sumsq[f], sq);
}

__global__ void k_bnfin(const float* __restrict__ sum, const float* __restrict__ sumsq,
                        const float* __restrict__ gamma, const float* __restrict__ beta,
                        float* __restrict__ scale, float* __restrict__ shift, int N) {
  int f = threadIdx.x;
  if (f < FDIM) {
    float inv = 1.0f / (float)N;
    float mu = sum[f] * inv;
    float var = sumsq[f] * inv - mu * mu;
    float rstd = rsqrtf(var + 1e-5f);
    float sc = rstd * gamma[f];
    scale[f] = sc;
    shift[f] = beta[f] - mu * sc;
  }
}

// in-place normalize + LeakyReLU(0.01)
__global__ void k_bnapply(float* __restrict__ h, const float* __restrict__ scale,
                          const float* __restrict__ shift, long NF) {
  long i = (long)blockIdx.x * blockDim.x + threadIdx.x;
  long stride = (long)gridDim.x * blockDim.x;
  for (; i < NF; i += stride) {
    int f = (int)(i & 63);
    float v = h[i] * scale[f] + shift[f];
    h[i] = (v >= 0.0f) ? v : 0.01f * v;
  }
}

// pooled sums: sums[batch[i], f] += h2[i, f]  (one thread per (node, 4-feat))
__global__ void k_pool(const int* __restrict__ batch, const float* __restrict__ h2,
                       float* __restrict__ sums, long N16) {
  long i = (long)blockIdx.x * blockDim.x + threadIdx.x;
  long stride = (long)gridDim.x * blockDim.x;
  for (; i < N16; i += stride) {
    long node = i >> 4;
    int f4 = (int)(i & 15) << 2;
    int g = batch[node];
    const float4 hv = *(const float4*)(h2 + node * FDIM + f4);
    float* sp = sums + (long)g * FDIM + f4;
    atomicAdd(sp + 0, hv.x);
    atomicAdd(sp + 1, hv.y);
    atomicAdd(sp + 2, hv.z);
    atomicAdd(sp + 3, hv.w);
  }
}

__global__ void k_cnt(const int* __restrict__ batch, float* __restrict__ cnts, int N) {
  int i = blockIdx.x * blockDim.x + threadIdx.x;
  int stride = gridDim.x * blockDim.x;
  for (; i < N; i += stride) atomicAdd(&cnts[batch[i]], 1.0f);
}

// out[g] = (sums[g,:]/max(cnt,1)) . linW + linb
__global__ void k_final(const float* __restrict__ sums, const float* __restrict__ cnts,
                        const float* __restrict__ linW, const float* __restrict__ linb,
                        float* __restrict__ out, int G) {
  int g = blockIdx.x * blockDim.x + threadIdx.x;
  if (g < G) {
    float c = cnts[g];
    if (c < 1.0f) c = 1.0f;
    float inv = 1.0f / c;
    float acc = linb[0];
    for (int f = 0; f < FDIM; ++f) acc += sums[(long)g * FDIM + f] * inv * linW[f];
    out[g] = acc;
  }
}

extern "C" void kernel_launch(void* const* d_in, const int* in_sizes, int n_in,
                              void* d_out, int out_size, void* d_ws, size_t ws_size,
                              hipStream_t stream) {
  const float* x     = (const float*)d_in[0];
  const int*   ei    = (const int*)d_in[1];
  const int*   batch = (const int*)d_in[2];
  const float* W1    = (const float*)d_in[3];   // (2,64,64)
  const float* b1    = (const float*)d_in[4];
  const float* W2    = (const float*)d_in[5];   // (2,64,64)
  const float* b2    = (const float*)d_in[6];
  const float* gamma = (const float*)d_in[7];
  const float* beta  = (const float*)d_in[8];
  const float* linW  = (const float*)d_in[9];
  const float* linb  = (const float*)d_in[10];
  float* out = (float*)d_out;

  const int N = in_sizes[0] / FDIM;
  const int E = in_sizes[1] / 2;
  const int G = out_size;
  const int* row = ei;
  const int* col = ei + E;
  const long N64 = (long)N * FDIM;
  const long E16 = (long)E * 16;
  const long N16 = (long)N * 16;

  // workspace layout (floats)
  float* ws    = (float*)d_ws;
  float* dis   = ws;                    // N
  float* w     = dis + N;               // E
  float* T     = w + E;                 // N*64 (reused for both layers)
  float* h     = T + N64;               // N*64
  float* h2    = h + N64;               // N*64
  float* sums  = h2 + N64;              // G*64
  float* cnts  = sums + (long)G * 64;   // G
  float* ssum  = cnts + G;              // 64
  float* ssq   = ssum + 64;             // 64
  float* scale = ssq + 64;              // 64
  float* shift = scale + 64;            // 64

  auto nblk = [](long n) {
    long b = (n + 255) / 256;
    return (int)(b > 65535 ? 65535 : b);
  };

  // ---- edge weights ----
  k_zero  <<<nblk(N),  256, 0, stream>>>(dis, N);
  k_deg   <<<nblk(E),  256, 0, stream>>>(row, dis, E);
  k_dis   <<<nblk(N),  256, 0, stream>>>(dis, N);
  k_edgew <<<nblk(E),  256, 0, stream>>>(row, col, dis, w, E);

  // ---- layer 1: Tx1 scatter + GEMM ----
  k_zero    <<<nblk(N64), 256, 0, stream>>>(T, N64);
  k_scatter <<<nblk(E16), 256, 0, stream>>>(row, col, w, x, T, E16);
  int gemmBlocks = (N + 127) / 128;
  k_cheb_gemm<<<gemmBlocks, 256, 0, stream>>>(x, T, W1, b1, h, N);

  // ---- BatchNorm + LeakyReLU ----
  k_zero    <<<1, 256, 0, stream>>>(ssum, 128);
  k_bnstats <<<(N + 255) / 256, 256, 0, stream>>>(h, ssum, ssq, N);
  k_bnfin   <<<1, 64, 0, stream>>>(ssum, ssq, gamma, beta, scale, shift, N);
  k_bnapply <<<nblk(N64), 256, 0, stream>>>(h, scale, shift, N64);

  // ---- layer 2: Tx scatter + GEMM ----
  k_zero    <<<nblk(N64), 256, 0, stream>>>(T, N64);
  k_scatter <<<nblk(E16), 256, 0, stream>>>(row, col, w, h, T, E16);
  k_cheb_gemm<<<gemmBlocks, 256, 0, stream>>>(h, T, W2, b2, h2, N);

  // ---- global mean pool + linear ----
  k_zero  <<<nblk((long)G * 64 + G), 256, 0, stream>>>(sums, (long)G * 64 + G);
  k_pool  <<<nblk(N16), 256, 0, stream>>>(batch, h2, sums, N16);
  k_cnt   <<<nblk(N),   256, 0, stream>>>(batch, cnts, N);
  k_final <<<(G + 63) / 64, 64, 0, stream>>>(sums, cnts, linW, linb, out, G);
}